// SmoothEdgeTermLossModule_21981642621254
// MI455X (gfx1250) — compile-verified
//
#include <hip/hip_runtime.h>

// Problem constants (match reference)
#define IMG_H 384
#define IMG_W 384
#define NCH   3
#define NB    4
#define HALO  3          // WINDOW // 2
#define TX    32
#define TY    8
#define LW    (TX + 2*HALO)   // 38
#define LH    (TY + 2*HALO)   // 14
#define LSTR  40              // padded row stride (floats) for bank friendliness

#define LOG2E 1.4426950408889634f

typedef __attribute__((ext_vector_type(2))) float v2f;
typedef __attribute__((ext_vector_type(8))) float v8f;

__device__ __forceinline__ float fast_exp2(float x) { return __builtin_amdgcn_exp2f(x); } // v_exp_f32
__device__ __forceinline__ float fast_log2(float x) { return __builtin_amdgcn_logf(x); }  // v_log_f32

__device__ __forceinline__ float lp04(float d) {
    // (d*d + 1e-6)^0.4  ==  exp2(0.4 * log2(d*d + 1e-6))
    return fast_exp2(0.4f * fast_log2(__fmaf_rn(d, d, 1e-6f)));
}

__global__ void zero_out_kernel(float* out) { out[0] = 0.0f; }

__global__ __launch_bounds__(256)
void smooth_edge_loss_kernel(const float* __restrict__ inp,
                             const float* __restrict__ tgt,
                             float* __restrict__ out)
{
    __shared__ float s_in[NCH][LH][LSTR];
    __shared__ float s_tg[NCH][LH][LSTR];
    __shared__ float s_red[8];

    const int tid   = threadIdx.x;
    const int tileX = blockIdx.x * TX;
    const int tileY = blockIdx.y * TY;
    const int b     = blockIdx.z;

    const size_t plane = (size_t)IMG_H * IMG_W;
    const float* ib = inp + (size_t)b * NCH * plane;
    const float* tb = tgt + (size_t)b * NCH * plane;

    // ---- cooperative haloed tile load with edge clamp (matches pad mode='edge')
    for (int idx = tid; idx < LW * LH; idx += 256) {
        const int lx = idx % LW;
        const int ly = idx / LW;
        int gx = tileX + lx - HALO;  gx = gx < 0 ? 0 : (gx > IMG_W - 1 ? IMG_W - 1 : gx);
        int gy = tileY + ly - HALO;  gy = gy < 0 ? 0 : (gy > IMG_H - 1 ? IMG_H - 1 : gy);
        const size_t g = (size_t)gy * IMG_W + gx;
#pragma unroll
        for (int c = 0; c < NCH; ++c) {
            s_in[c][ly][lx] = ib[c * plane + g];
            s_tg[c][ly][lx] = tb[c * plane + g];
        }
    }
    __syncthreads();

    const int tx = tid & 31;
    const int ty = tid >> 5;
    const int lx = tx + HALO;
    const int ly = ty + HALO;

    float ic[NCH], tc[NCH];
#pragma unroll
    for (int c = 0; c < NCH; ++c) { ic[c] = s_in[c][ly][lx]; tc[c] = s_tg[c][ly][lx]; }

    // L2 term (W_L2 = 1)
    float acc = 0.0f;
#pragma unroll
    for (int c = 0; c < NCH; ++c) { const float d = ic[c] - tc[c]; acc = __fmaf_rn(d, d, acc); }

    // spatial Gaussian table: es[k] = exp(-k^2 / (2 * 7^2)), separable
    const float neg_inv2ss_l2e = -(1.0f / (2.0f * 7.0f * 7.0f)) * LOG2E;
    float es[HALO + 1];
#pragma unroll
    for (int k = 0; k <= HALO; ++k) es[k] = fast_exp2((float)(k * k) * neg_inv2ss_l2e);

    // wc = exp(-cd / (2*0.1^2)) = exp2(cd * (-50 * log2 e))
    const float negc = -50.0f * LOG2E;

#pragma unroll
    for (int dy = -HALO; dy <= HALO; ++dy) {
        const float ey = es[dy < 0 ? -dy : dy];
#pragma unroll
        for (int dx = -HALO; dx <= HALO; ++dx) {
            if (dy == 0 && dx == 0) continue;
            const float ws = ey * es[dx < 0 ? -dx : dx];
            float cd = 0.0f, sm = 0.0f, ed = 0.0f;
#pragma unroll
            for (int c = 0; c < NCH; ++c) {
                const float iq = s_in[c][ly + dy][lx + dx];
                const float tq = s_tg[c][ly + dy][lx + dx];
                const float dt = tc[c] - tq;
                const float di = ic[c] - iq;
                cd = __fmaf_rn(dt, dt, cd);
                sm += lp04(di);
                const float ad = fabsf(di) - fabsf(dt);
                ed += lp04(ad);
            }
            const float wc = fast_exp2(cd * negc);
            // W_SMOOTH = W_EDGE = 1:  wc*sm + (1-wc)*ed == fma(wc, sm-ed, ed)
            acc = __fmaf_rn(ws, __fmaf_rn(wc, sm - ed, ed), acc);
        }
    }

    // ---- wave32 reduction via V_WMMA_F32_16X16X4_F32:
    // A (16x4) holds lane partials: A[m][:] = {v[m], 0, v[16+m], 0}; B = ones.
    // D[m][n] = v[m] + v[16+m] replicated across N. Sum D's 8 VGPRs per lane,
    // then one shfl_xor(16) yields the full 32-lane sum in every lane.
    v2f a;  a[0] = acc;  a[1] = 0.0f;
    v2f bm; bm[0] = 1.0f; bm[1] = 1.0f;
    v8f c8 = {};
    v8f d8 = __builtin_amdgcn_wmma_f32_16x16x4_f32(false, a, false, bm,
                                                   (short)0, c8, false, false);
    float s = 0.0f;
#pragma unroll
    for (int i = 0; i < 8; ++i) s += d8[i];
    s += __shfl_xor(s, 16, 32);

    if (tx == 0) s_red[ty] = s;
    __syncthreads();
    if (tid == 0) {
        float t = 0.0f;
#pragma unroll
        for (int w = 0; w < 8; ++w) t += s_red[w];
        const float inv_n = 1.0f / (float)((size_t)NB * NCH * IMG_H * IMG_W);
        atomicAdd(out, t * inv_n);
    }
}

extern "C" void kernel_launch(void* const* d_in, const int* in_sizes, int n_in,
                              void* d_out, int out_size, void* d_ws, size_t ws_size,
                              hipStream_t stream)
{
    (void)in_sizes; (void)n_in; (void)d_ws; (void)ws_size; (void)out_size;
    const float* inp = (const float*)d_in[0];
    const float* tgt = (const float*)d_in[1];
    float* out = (float*)d_out;

    zero_out_kernel<<<1, 1, 0, stream>>>(out);

    dim3 grid(IMG_W / TX, IMG_H / TY, NB);   // 12 x 48 x 4 = 2304 blocks
    smooth_edge_loss_kernel<<<grid, 256, 0, stream>>>(inp, tgt, out);
}